// CosineSimilarity_5162550689872
// MI455X (gfx1250) — compile-verified
//
#include <hip/hip_runtime.h>
#include <hip/hip_bf16.h>

// ---------------------------------------------------------------------------
// Cosine distance over 5 fp32 tensor pairs: memory-bound streaming reduction.
// ~365MB read once @ 23.3 TB/s -> ~16us ideal. Main loop: b128 non-temporal
// loads + dual FMAC. Wave-level reduction done on the WMMA engine
// (v_wmma_f32_16x16x4_f32 with a ones-B matrix acts as a cross-lane adder).
// Two-phase deterministic reduction (no float atomics).
// ---------------------------------------------------------------------------

typedef __attribute__((ext_vector_type(2))) float v2f;
typedef __attribute__((ext_vector_type(4))) float v4f;
typedef __attribute__((ext_vector_type(8))) float v8f;

// One WMMA: with A.v0 = x, A.v1 = 0, B = ones(4x16):
//   D[m][n] = A[m][0] + A[m][2] = x_m + x_{m+16}
// Summing the 8 D VGPRs per lane gives half-wave totals:
//   lanes 0-15 : sum over m=0..7  of (x_m + x_{m+16})
//   lanes 16-31: sum over m=8..15 of (x_m + x_{m+16})
__device__ __forceinline__ float wmma_halfwave_sum(float x) {
    v2f a; a[0] = x;    a[1] = 0.0f;
    v2f b; b[0] = 1.0f; b[1] = 1.0f;
    v8f c = {};
    v8f d = __builtin_amdgcn_wmma_f32_16x16x4_f32(
        /*neg_a=*/false, a, /*neg_b=*/false, b,
        /*c_mod=*/(short)0, c, /*reuse_a=*/false, /*reuse_b=*/false);
    float s01 = d[0] + d[1], s23 = d[2] + d[3];
    float s45 = d[4] + d[5], s67 = d[6] + d[7];
    return (s01 + s23) + (s45 + s67);
}

__global__ __launch_bounds__(256) void cosred_partial(
    const float* __restrict__ r0, const float* __restrict__ r1,
    const float* __restrict__ r2, const float* __restrict__ r3,
    const float* __restrict__ r4,
    const float* __restrict__ d0, const float* __restrict__ d1p,
    const float* __restrict__ d2p, const float* __restrict__ d3p,
    const float* __restrict__ d4p,
    int n0, int n1, int n2, int n3, int n4,
    float* __restrict__ ws)
{
    const float* recs[5] = {r0, r1, r2, r3, r4};
    const float* dats[5] = {d0, d1p, d2p, d3p, d4p};
    const int    ns[5]   = {n0, n1, n2, n3, n4};

    // 32-bit indices: all element counts < 2^31 and stride ~1M, so no overflow.
    const int tid    = (int)(blockIdx.x * blockDim.x + threadIdx.x);
    const int stride = (int)(gridDim.x * blockDim.x);

    float sp = 0.0f, rn = 0.0f, dn = 0.0f;

#pragma unroll
    for (int p = 0; p < 5; ++p) {
        const v4f* __restrict__ rv = (const v4f*)recs[p];
        const v4f* __restrict__ dv = (const v4f*)dats[p];
        const int nq = ns[p] >> 2;
        for (int i = tid; i < nq; i += stride) {
            // Streaming, zero-reuse data (365MB > 192MB L2): NT b128 loads.
            v4f r = __builtin_nontemporal_load(&rv[i]);
            v4f d = __builtin_nontemporal_load(&dv[i]);
            sp = fmaf(r[0], d[0], sp); sp = fmaf(r[1], d[1], sp);
            sp = fmaf(r[2], d[2], sp); sp = fmaf(r[3], d[3], sp);
            rn = fmaf(r[0], r[0], rn); rn = fmaf(r[1], r[1], rn);
            rn = fmaf(r[2], r[2], rn); rn = fmaf(r[3], r[3], rn);
            dn = fmaf(d[0], d[0], dn); dn = fmaf(d[1], d[1], dn);
            dn = fmaf(d[2], d[2], dn); dn = fmaf(d[3], d[3], dn);
        }
        // Tail (all sizes are multiples of 4, but stay safe).
        for (int i = (nq << 2) + tid; i < ns[p]; i += stride) {
            float r = recs[p][i], d = dats[p][i];
            sp = fmaf(r, d, sp);
            rn = fmaf(r, r, rn);
            dn = fmaf(d, d, dn);
        }
    }

    // ---- wave reduction on the matrix engine ----
    // Everything below is straight-line (no EXEC-divergent control flow)
    // until after the last WMMA: EXEC is all-1s as WMMA requires.
    const int lane = threadIdx.x & 31;
    const int wid  = threadIdx.x >> 5;

    float tsp = wmma_halfwave_sum(sp);
    float trn = wmma_halfwave_sum(rn);
    float tdn = wmma_halfwave_sum(dn);

    // Stage 2: lanes {0,16} carry tsp, {1,17} trn, {2,18} tdn; B = ones.
    // D2[0][n] = tsp_0 + tsp_16 = wave sum(sp); rows 1/2 likewise for rn/dn.
    // Branchless select -> v_cndmask + v_fma, keeps EXEC untouched.
    const int l16 = lane & 15;
    const float m0 = (l16 == 0) ? 1.0f : 0.0f;
    const float m1 = (l16 == 1) ? 1.0f : 0.0f;
    const float m2 = (l16 == 2) ? 1.0f : 0.0f;
    float sel = fmaf(tsp, m0, fmaf(trn, m1, tdn * m2));

    v2f a2; a2[0] = sel;  a2[1] = 0.0f;
    v2f b2; b2[0] = 1.0f; b2[1] = 1.0f;
    v8f c2 = {};
    v8f d2 = __builtin_amdgcn_wmma_f32_16x16x4_f32(
        false, a2, false, b2, (short)0, c2, false, false);
    // lanes 0-15: d2[0]=wave sum(sp), d2[1]=wave sum(rn), d2[2]=wave sum(dn)

    __shared__ float wsum[8][3];
    if (lane == 0) {
        wsum[wid][0] = d2[0];
        wsum[wid][1] = d2[1];
        wsum[wid][2] = d2[2];
    }
    __syncthreads();

    if (threadIdx.x < 3) {
        float s = 0.0f;
        const int nwaves = blockDim.x >> 5;
        for (int w = 0; w < nwaves; ++w) s += wsum[w][threadIdx.x];
        ws[blockIdx.x * 3 + threadIdx.x] = s;
    }
}

__global__ __launch_bounds__(256) void cosred_final(
    const float* __restrict__ ws, int nblocks, float* __restrict__ out)
{
    __shared__ float buf[256];
    __shared__ float tot[3];
    const int t = threadIdx.x;
    const int c = t & 3;

    float s = 0.0f;
    if (c < 3) {
        for (int i = (t >> 2); i < nblocks; i += 64)
            s += ws[i * 3 + c];
    }
    buf[t] = s;
    __syncthreads();

    if (t < 3) {
        float acc = 0.0f;
        for (int j = 0; j < 64; ++j) acc += buf[t + (j << 2)];
        tot[t] = acc;
    }
    __syncthreads();

    if (t == 0) {
        // 1 - sp / (sqrt(rn) * sqrt(dn))
        out[0] = 1.0f - tot[0] / (sqrtf(tot[1]) * sqrtf(tot[2]));
    }
}

extern "C" void kernel_launch(void* const* d_in, const int* in_sizes, int n_in,
                              void* d_out, int out_size, void* d_ws, size_t ws_size,
                              hipStream_t stream) {
    (void)n_in; (void)out_size;
    const float* r[5];
    const float* d[5];
    int n[5];
    for (int i = 0; i < 5; ++i) {
        r[i] = (const float*)d_in[i];
        d[i] = (const float*)d_in[i + 5];
        n[i] = in_sizes[i];
    }

    int blocks = 4096;                               // 32K wave32s in flight
    const size_t per_block = 3 * sizeof(float);
    if ((size_t)blocks * per_block > ws_size) {
        blocks = (int)(ws_size / per_block);
        if (blocks < 1) blocks = 1;
    }

    float* ws = (float*)d_ws;
    cosred_partial<<<blocks, 256, 0, stream>>>(
        r[0], r[1], r[2], r[3], r[4],
        d[0], d[1], d[2], d[3], d[4],
        n[0], n[1], n[2], n[3], n[4],
        ws);
    cosred_final<<<1, 256, 0, stream>>>(ws, blocks, (float*)d_out);
}